// GRULayer_39118562132278
// MI455X (gfx1250) — compile-verified
//
#include <hip/hip_runtime.h>
#include <hip/hip_bf16.h>

// ---------------------------------------------------------------------------
// 2-layer GRU (PyTorch gate order r,z,n), B=128, T=1024, IN=256, H=128, fp32.
//   1) gx0 = x  @ w_ih0^T + b_ih0        (WMMA f32 GEMM, [131072 x 384])
//   2) h1  = scan(gx0, w_hh0, b_hh0)     (batch-sliced scan, TDM-fed LDS)
//   3) gx1 = h1 @ w_ih1^T + b_ih1
//   4) out = scan(gx1, w_hh1, b_hh1)
// ---------------------------------------------------------------------------

typedef __attribute__((ext_vector_type(2))) float v2f;
typedef __attribute__((ext_vector_type(8))) float v8f;
typedef __attribute__((ext_vector_type(4))) unsigned int u32x4;
typedef __attribute__((ext_vector_type(8))) int i32x8;
typedef __attribute__((ext_vector_type(4))) int i32x4;

#define GRU_B   128
#define GRU_T   1024
#define GRU_IN  256
#define GRU_H   128
#define GRU_G3  384   // 3*H

// Padded LDS row strides (floats) -> conflict-free b64 LDS accesses:
// bank = (stride%64 * row + col) % 64 ; stride%64==4 interleaves half-waves.
#define SW_LD   132   // w_hh rows (128 + 4)
#define SH_LD   132   // h rows
#define SGH_LD  388   // gh rows (384 + 4)

// ---------------------------------------------------------------------------
// Tensor Data Mover: DMA a 2D tile (width x rows of f32, row stride in elts)
// from global memory into LDS at byte offset lds_off. Optional LDS padding
// (pad_amount+1 DWORDs inserted every 2^(pad_interval+1) DWORDs stored).
// Wave-level op (EXEC ignored); tracked by TENSORcnt. ISA §8.3/§8.4.
// ---------------------------------------------------------------------------
__device__ __forceinline__ void tdm_load_2d_f32(const float* gptr,
                                                unsigned lds_off,
                                                unsigned width_elts,
                                                unsigned rows,
                                                unsigned stride_elts,
                                                unsigned pad_enable,
                                                unsigned pad_interval,
                                                unsigned pad_amount) {
  unsigned long long ga = (unsigned long long)(const void*)gptr;
  u32x4 g0;
  g0[0] = 1u;                                         // count=1, user D#
  g0[1] = lds_off;                                    // lds_addr (bytes)
  g0[2] = (unsigned)(ga & 0xFFFFFFFFu);               // global_addr[31:0]
  g0[3] = (unsigned)((ga >> 32) & 0x01FFFFFFu)        // global_addr[56:32]
          | (2u << 30);                               // type = 2 ("image")
  i32x8 g1;
  g1[0] = (int)((2u << 16) |                          // data_size=2 (4 bytes)
                (pad_enable << 20) |
                (pad_interval << 22) |
                (pad_amount << 25));
  g1[1] = (int)((width_elts & 0xFFFFu) << 16);        // tensor_dim0[15:0]
  g1[2] = (int)((width_elts >> 16) |
                ((rows & 0xFFFFu) << 16));            // dim0[31:16] | dim1[15:0]
  g1[3] = (int)((rows >> 16) |
                ((width_elts & 0xFFFFu) << 16));      // dim1[31:16] | tile_dim0
  g1[4] = (int)(rows & 0xFFFFu);                      // tile_dim1 (tile_dim2=0)
  g1[5] = (int)stride_elts;                           // tensor_dim0_stride lo32
  g1[6] = 0;                                          // stride hi16 | dim1_stride
  g1[7] = 0;
  i32x4 z4 = {0, 0, 0, 0};                            // groups 2/3 unused (2D)
  i32x8 z8 = {0, 0, 0, 0, 0, 0, 0, 0};
  __builtin_amdgcn_tensor_load_to_lds(g0, g1, z4, z4, z8, 0);
}

__device__ __forceinline__ unsigned lds_offset_of(const void* p) {
  // addrspacecast(LDS->flat) keeps the workgroup-relative offset in addr[31:0]
  return (unsigned)(unsigned long long)p;
}

__device__ __forceinline__ float fast_sigmoid(float x) {
  return 1.0f / (1.0f + __expf(-x));
}
__device__ __forceinline__ float fast_tanh(float x) {
  float e = __expf(-2.0f * x);
  return (1.0f - e) / (1.0f + e);
}

// ---------------------------------------------------------------------------
// GEMM: out[M,N] = X[M,K] @ W[N,K]^T + bias[N]
// Each wave computes a 64x16 strip: 4 M-tiles reuse every B fragment.
// Two-stage ping-pong software pipeline (no register-rotation movs).
// V_WMMA_F32_16X16X4_F32; A frag: lane L -> m=L&15, K={kb,kb+1}, kb=(L>>4)*2;
// B frag: lane L -> n=L&15, same K pair; C/D: vgpr v -> m=v+(L>>4)*8, n=L&15.
// ---------------------------------------------------------------------------
#define GEMM_LOAD(SB, S0, S1, S2, S3, OFF)  \
  SB = *(const v2f*)&W[brow + (OFF)];       \
  S0 = *(const v2f*)&X[a0 + (OFF)];         \
  S1 = *(const v2f*)&X[a1 + (OFF)];         \
  S2 = *(const v2f*)&X[a2 + (OFF)];         \
  S3 = *(const v2f*)&X[a3 + (OFF)];

#define GEMM_MMA(SB, S0, S1, S2, S3)                                           \
  acc0 = __builtin_amdgcn_wmma_f32_16x16x4_f32(false, S0, false, SB, (short)0, \
                                               acc0, false, false);            \
  acc1 = __builtin_amdgcn_wmma_f32_16x16x4_f32(false, S1, false, SB, (short)0, \
                                               acc1, false, false);            \
  acc2 = __builtin_amdgcn_wmma_f32_16x16x4_f32(false, S2, false, SB, (short)0, \
                                               acc2, false, false);            \
  acc3 = __builtin_amdgcn_wmma_f32_16x16x4_f32(false, S3, false, SB, (short)0, \
                                               acc3, false, false);

__global__ __launch_bounds__(256) void gru_gx_gemm(
    const float* __restrict__ X, const float* __restrict__ W,
    const float* __restrict__ bias, float* __restrict__ out,
    int M, int K, int N) {
  const int lane = threadIdx.x & 31;
  const int wave = threadIdx.x >> 5;
  const int tiles_n = N >> 4;

  const long gwave  = (long)blockIdx.x * (blockDim.x >> 5) + wave;
  const long strip  = gwave / tiles_n;       // 64-row strip index
  const int  tile_n = (int)(gwave % tiles_n);
  if (strip * 64 >= M) return;               // wave-uniform exit

  const int mr = lane & 15;
  const int kb = (lane >> 4) << 1;
  const long a0 = (strip * 64 + mr) * (long)K;
  const long a1 = a0 + 16 * (long)K;
  const long a2 = a0 + 32 * (long)K;
  const long a3 = a0 + 48 * (long)K;
  const long brow = (long)(tile_n * 16 + mr) * K;

  v8f acc0 = {}, acc1 = {}, acc2 = {}, acc3 = {};
  v2f bA, p0, p1, p2, p3;    // ping fragment set
  v2f bB, q0, q1, q2, q3;    // pong fragment set

  GEMM_LOAD(bA, p0, p1, p2, p3, kb)          // k = 0
  int k = 0;
  for (; k + 8 < K; k += 8) {                // K is a multiple of 8
    GEMM_LOAD(bB, q0, q1, q2, q3, k + 4 + kb)
    GEMM_MMA(bA, p0, p1, p2, p3)
    GEMM_LOAD(bA, p0, p1, p2, p3, k + 8 + kb)
    GEMM_MMA(bB, q0, q1, q2, q3)
  }
  GEMM_LOAD(bB, q0, q1, q2, q3, k + 4 + kb)  // k == K-8 here
  GEMM_MMA(bA, p0, p1, p2, p3)
  GEMM_MMA(bB, q0, q1, q2, q3)

  const int n = tile_n * 16 + mr;
  const float bn_ = bias[n];
  const long m0 = strip * 64 + ((lane >> 4) << 3);
#pragma unroll
  for (int v = 0; v < 8; ++v) out[(m0 + v) * (long)N + n]      = acc0[v] + bn_;
#pragma unroll
  for (int v = 0; v < 8; ++v) out[(m0 + 16 + v) * (long)N + n] = acc1[v] + bn_;
#pragma unroll
  for (int v = 0; v < 8; ++v) out[(m0 + 32 + v) * (long)N + n] = acc2[v] + bn_;
#pragma unroll
  for (int v = 0; v < 8; ++v) out[(m0 + 48 + v) * (long)N + n] = acc3[v] + bn_;
}

// ---------------------------------------------------------------------------
// Recurrent scan. Each block (256 thr) owns 16 batch rows for all T steps.
// LDS: w_hh staged once via TDM into a bank-conflict-free padded layout;
// gx_t slices double-buffered via TDM issued 2 steps ahead; gh/h resident.
// All t-invariant addresses are precomputed outside the serial loop.
// ---------------------------------------------------------------------------
__global__ __launch_bounds__(256) void gru_scan(
    const float* __restrict__ gx, const float* __restrict__ w_hh,
    const float* __restrict__ b_hh, float* __restrict__ out, int T) {
  extern __shared__ float smem[];
  float* s_w   = smem;                         // 384 x SW_LD (padded rows)
  float* s_gh  = s_w + GRU_G3 * SW_LD;         // 16 x SGH_LD
  float* s_h   = s_gh + 16 * SGH_LD;           // 16 x SH_LD
  float* s_b   = s_h + 16 * SH_LD;             // 384
  float* s_gx0 = s_b + GRU_G3;                 // 16*384 (gx double-buffer 0)
  float* s_gx1 = s_gx0 + 16 * GRU_G3;          // 16*384 (gx double-buffer 1)

  const int tid  = threadIdx.x;
  const int lane = tid & 31;
  const int wave = tid >> 5;
  const int b0   = blockIdx.x * 16;
  const long row0 = (long)b0 * T;              // gx row of (b0, t=0)
  const unsigned gx_stride = (unsigned)T * GRU_G3;

  if (wave == 0) {
    // w_hh as 384 rows x 128 elts, LDS-padded +4 DWORDs per 128-DWORD row
    // (pad_interval=6 -> 128 DWORDs, pad_amount=3 -> 4 DWORDs) => stride 132.
    tdm_load_2d_f32(w_hh, lds_offset_of(s_w), GRU_H, GRU_G3, GRU_H,
                    1u, 6u, 3u);
    tdm_load_2d_f32(gx + row0 * GRU_G3, lds_offset_of(s_gx0),
                    GRU_G3, 16, gx_stride, 0u, 0u, 0u);
    if (T > 1)
      tdm_load_2d_f32(gx + (row0 + 1) * GRU_G3, lds_offset_of(s_gx1),
                      GRU_G3, 16, gx_stride, 0u, 0u, 0u);
    __builtin_amdgcn_s_wait_tensorcnt(2);      // in-order: w_hh has landed
  }
  for (int i = tid; i < GRU_G3; i += blockDim.x)     s_b[i] = b_hh[i];
  for (int i = tid; i < 16 * SH_LD; i += blockDim.x) s_h[i] = 0.0f;
  __syncthreads();

  const int mr = lane & 15;
  const int kb = (lane >> 4) << 1;
  const int r0 = (lane >> 4) << 3;

  // t-invariant per-wave WMMA state: column indices + biases of the 3 tiles.
  int   ncol[3];
  float bncol[3];
#pragma unroll
  for (int tt = 0; tt < 3; ++tt) {
    ncol[tt]  = (wave * 3 + tt) * 16 + mr;
    bncol[tt] = s_b[ncol[tt]];
  }

  // t-invariant gate-phase state: each thread owns 8 (m,j) elements.
  unsigned oxg[8], ogh[8], oh[8];
  float*   outp[8];
#pragma unroll
  for (int q = 0; q < 8; ++q) {
    const int i = tid + q * 256;               // 0..2047
    const int m = i >> 7;
    const int j = i & (GRU_H - 1);
    oxg[q] = m * GRU_G3 + j;
    ogh[q] = m * SGH_LD + j;
    oh[q]  = m * SH_LD + j;
    outp[q] = out + (long)(b0 + m) * T * GRU_H + j;   // step stride = GRU_H
  }

  for (int t = 0; t < T; ++t) {
    // ---- gh = h @ w_hh^T + b_hh : A-frags cached once, 3 N-tiles/wave -----
    v2f afr[32];
#pragma unroll
    for (int kk = 0; kk < 32; ++kk)
      afr[kk] = *(const v2f*)&s_h[mr * SH_LD + (kk << 2) + kb];
#pragma unroll
    for (int tt = 0; tt < 3; ++tt) {
      const int n = ncol[tt];
      v8f acc = {};
#pragma unroll
      for (int kk = 0; kk < 32; ++kk) {
        v2f b = *(const v2f*)&s_w[n * SW_LD + (kk << 2) + kb];
        acc = __builtin_amdgcn_wmma_f32_16x16x4_f32(
            false, afr[kk], false, b, (short)0, acc, false, false);
      }
      const float bn = bncol[tt];
#pragma unroll
      for (int v = 0; v < 8; ++v) s_gh[(r0 + v) * SGH_LD + n] = acc[v] + bn;
    }
    // gx slice for step t must have landed before the gate phase.
    if (wave == 0) {
      if (t + 1 < T) __builtin_amdgcn_s_wait_tensorcnt(1);
      else           __builtin_amdgcn_s_wait_tensorcnt(0);
    }
    __syncthreads();   // s_gh complete + gx buffer[t&1] valid for all waves

    // ---- gates + h update: 8 owned elements per thread --------------------
    const float* s_gx = (t & 1) ? s_gx1 : s_gx0;
#pragma unroll
    for (int q = 0; q < 8; ++q) {
      const float gxr = s_gx[oxg[q]];
      const float gxz = s_gx[oxg[q] + GRU_H];
      const float gxn = s_gx[oxg[q] + 2 * GRU_H];
      const float ghr = s_gh[ogh[q]];
      const float ghz = s_gh[ogh[q] + GRU_H];
      const float ghn = s_gh[ogh[q] + 2 * GRU_H];
      const float r  = fast_sigmoid(gxr + ghr);
      const float z  = fast_sigmoid(gxz + ghz);
      const float nn = fast_tanh(gxn + r * ghn);
      const float hp = s_h[oh[q]];
      const float hn = (1.0f - z) * nn + z * hp;
      s_h[oh[q]] = hn;
      *outp[q] = hn;
      outp[q] += GRU_H;
    }
    __syncthreads();   // everyone done with gx buffer[t&1] and s_h updated

    // Refill the buffer just consumed with the slice for step t+2.
    if (wave == 0 && t + 2 < T) {
      float* dst = (t & 1) ? s_gx1 : s_gx0;
      tdm_load_2d_f32(gx + (row0 + t + 2) * GRU_G3, lds_offset_of(dst),
                      GRU_G3, 16, gx_stride, 0u, 0u, 0u);
    }
  }
}

// ---------------------------------------------------------------------------
extern "C" void kernel_launch(void* const* d_in, const int* in_sizes, int n_in,
                              void* d_out, int out_size, void* d_ws, size_t ws_size,
                              hipStream_t stream) {
  const float* x     = (const float*)d_in[0];
  const float* w_ih0 = (const float*)d_in[1];
  const float* w_hh0 = (const float*)d_in[2];
  const float* b_ih0 = (const float*)d_in[3];
  const float* b_hh0 = (const float*)d_in[4];
  const float* w_ih1 = (const float*)d_in[5];
  const float* w_hh1 = (const float*)d_in[6];
  const float* b_ih1 = (const float*)d_in[7];
  const float* b_hh1 = (const float*)d_in[8];
  float* out = (float*)d_out;

  const int M = GRU_B * GRU_T;  // 131072
  float* ws_gx = (float*)d_ws;                   // [M x 384]
  float* ws_h1 = ws_gx + (size_t)M * GRU_G3;     // [M x 128]

  const size_t scan_lds =
      (GRU_G3 * SW_LD +      // w_hh (padded)
       16 * SGH_LD +         // gh   (padded)
       16 * SH_LD +          // h    (padded)
       GRU_G3 +              // b_hh
       2 * 16 * GRU_G3) *    // gx double buffers
      sizeof(float);         // = 286,720 B < 320 KB

  // Each wave covers a 64x16 strip; 8 waves per 256-thread block.
  const int strips = (M / 64) * (GRU_G3 / 16);
  const int gemm_blocks = strips / 8;

  // Layer 0
  gru_gx_gemm<<<gemm_blocks, 256, 0, stream>>>(x, w_ih0, b_ih0, ws_gx,
                                               M, GRU_IN, GRU_G3);
  gru_scan<<<GRU_B / 16, 256, scan_lds, stream>>>(ws_gx, w_hh0, b_hh0,
                                                  ws_h1, GRU_T);
  // Layer 1
  gru_gx_gemm<<<gemm_blocks, 256, 0, stream>>>(ws_h1, w_ih1, b_ih1, ws_gx,
                                               M, GRU_H, GRU_G3);
  gru_scan<<<GRU_B / 16, 256, scan_lds, stream>>>(ws_gx, w_hh1, b_hh1,
                                                  out, GRU_T);
}